// DictPlenoxels_84061099917635
// MI455X (gfx1250) — compile-verified
//
#include <hip/hip_runtime.h>

// ---------------------------------------------------------------------------
// DictPlenoxels forward, MI455X / gfx1250.
//
// Key algebraic optimization: output only uses data[:, -1], i.e. per corner
//   dot(grid_row[32], atoms_last[fine_cell][32])
// with only 8 possible fine cells. Precompute S[32768, 8] = grid @ AL once
// with V_WMMA_F32_16X16X4_F32 (exact fp32), then the per-sample loop is just
// 8 scalar gathers + trilinear weights.
// ---------------------------------------------------------------------------

typedef __attribute__((ext_vector_type(2))) float v2f;
typedef __attribute__((ext_vector_type(8))) float v8f;

#define NATOMS   32
#define DDIM     13
#define NCELL    (32 * 32 * 32)     // 32768 coarse cells
#define GRES     64                 // COARSE * FINE
#define VOXEL_F  0.03125f           // 2R / 64
#define STEP_F   0.015625f          // VOXEL / 2
#define NSAMP    219                // N_INT - 1
#define NRAYS    1024

// ---------------------------------------------------------------------------
// Kernel 1: S[cell][f] = sum_a grid[cell][a] * atoms[f][a][12]
// GEMM [32768 x 32] @ [32 x 16(8 used)] via chained V_WMMA_F32_16X16X4_F32.
// 256 blocks x 8 waves = 2048 waves, one 16-row tile each.
// ---------------------------------------------------------------------------
__global__ __launch_bounds__(256) void dp_precompute_S(
    const float* __restrict__ grid,   // [32768][32]
    const float* __restrict__ atoms,  // [8][32][13]
    float* __restrict__ S)            // [32768][8]
{
    __shared__ float Bmat[32][16];    // Bmat[a][f] = atoms[(f*32+a)*13 + 12], f<8 else 0

    const int tid = threadIdx.x;
    for (int i = tid; i < 32 * 16; i += 256) {
        const int a = i >> 4;
        const int f = i & 15;
        Bmat[a][f] = (f < 8) ? atoms[(f * NATOMS + a) * DDIM + (DDIM - 1)] : 0.0f;
    }
    __syncthreads();

    const int lane  = tid & 31;
    const int wave  = tid >> 5;       // 0..7
    const int m     = lane & 15;      // M index (A rows / D cols)
    const int khalf = lane >> 4;      // 0 or 1

    // Loop-invariant B fragments. ISA layout (f32 4x16 B, 2 VGPRs):
    //   VGPR j, lane L  holds  B[(L/16)*2 + j, L%16]
    v2f bfrag[8];
#pragma unroll
    for (int kc = 0; kc < 8; ++kc) {
        const int k0 = kc * 4 + khalf * 2;
        bfrag[kc].x = Bmat[k0 + 0][m];
        bfrag[kc].y = Bmat[k0 + 1][m];
    }

    const int tile = blockIdx.x * 8 + wave;        // 0..2047
    const float* __restrict__ arow = grid + (size_t)(tile * 16 + m) * NATOMS;

    // A layout (f32 16x4, 2 VGPRs): VGPR j, lane L holds A[L%16, (L/16)*2 + j]
    v8f c = {};
#pragma unroll
    for (int kc = 0; kc < 8; ++kc) {
        const int k0 = kc * 4 + khalf * 2;
        v2f afrag;
        afrag.x = arow[k0 + 0];
        afrag.y = arow[k0 + 1];
        c = __builtin_amdgcn_wmma_f32_16x16x4_f32(
                /*neg_a=*/false, afrag, /*neg_b=*/false, bfrag[kc],
                /*c_mod=*/(short)0, c, /*reuse_a=*/false, /*reuse_b=*/false);
    }

    // D layout: VGPR r, lane L holds D[r + (L/16)*8, L%16]. Keep columns 0..7.
    if (m < 8) {
#pragma unroll
        for (int r = 0; r < 8; ++r) {
            const int outrow = tile * 16 + r + khalf * 8;
            S[outrow * 8 + m] = c[r];
        }
    }
}

// ---------------------------------------------------------------------------
// Kernel 2: per (ray, sample) — trilinear corner weights + 8 gathers from S.
// ---------------------------------------------------------------------------
__global__ __launch_bounds__(256) void dp_render(
    const float* __restrict__ rays_o,  // [1024][3]
    const float* __restrict__ rays_d,  // [1024][3]
    const float* __restrict__ S,       // [32768][8]
    float* __restrict__ out)           // [1024][219]
{
    const int r = blockIdx.x;
    const int s = threadIdx.x;
    if (s >= NSAMP) return;

    const float ox = rays_o[r * 3 + 0], oy = rays_o[r * 3 + 1], oz = rays_o[r * 3 + 2];
    const float dx = rays_d[r * 3 + 0], dy = rays_d[r * 3 + 1], dz = rays_d[r * 3 + 2];

    // start = max over dims of min((R - o)/d, (-R - o)/d)
    const float tx = fminf((1.0f - ox) / dx, (-1.0f - ox) / dx);
    const float ty = fminf((1.0f - oy) / dy, (-1.0f - oy) / dy);
    const float tz = fminf((1.0f - oz) / dz, (-1.0f - oz) / dz);
    const float start = fmaxf(fmaxf(tx, ty), tz);

    const float t  = start + (float)s * STEP_F;
    const float px = ox + t * dx;
    const float py = oy + t * dy;
    const float pz = oz + t * dz;

    const bool mask = (px > -1.0f) && (px < 1.0f) &&
                      (py > -1.0f) && (py < 1.0f) &&
                      (pz > -1.0f) && (pz < 1.0f);

    const float gx = (px + 1.0f) * (1.0f / VOXEL_F);
    const float gy = (py + 1.0f) * (1.0f / VOXEL_F);
    const float gz = (pz + 1.0f) * (1.0f / VOXEL_F);

    float acc = 0.0f;
#pragma unroll
    for (int n = 0; n < 8; ++n) {
        // CORNERS order: bit2 -> x, bit1 -> y, bit0 -> z; values +-1 (halved)
        const float cx = (n & 4) ? 0.5f : -0.5f;
        const float cy = (n & 2) ? 0.5f : -0.5f;
        const float cz = (n & 1) ? 0.5f : -0.5f;

        float fx = fminf(fmaxf(floorf(gx + cx), 0.0f), (float)(GRES - 1));
        float fy = fminf(fmaxf(floorf(gy + cy), 0.0f), (float)(GRES - 1));
        float fz = fminf(fmaxf(floorf(gz + cz), 0.0f), (float)(GRES - 1));

        const float wx = 1.0f - fabsf(gx - (fx + 0.5f));
        const float wy = 1.0f - fabsf(gy - (fy + 0.5f));
        const float wz = 1.0f - fabsf(gz - (fz + 0.5f));
        const float w  = wx * wy * wz;

        const int ix = (int)fx, iy = (int)fy, iz = (int)fz;
        const int cell = ((ix >> 1) * 32 + (iy >> 1)) * 32 + (iz >> 1);
        const int f    = ((ix & 1) << 2) | ((iy & 1) << 1) | (iz & 1);

        acc = fmaf(w, S[cell * 8 + f], acc);
    }

    out[r * NSAMP + s] = mask ? fmaxf(acc, 0.0f) : 0.0f;
}

// ---------------------------------------------------------------------------
// Fallback (only if workspace is too small for S): direct 32-deep dot.
// ---------------------------------------------------------------------------
__global__ __launch_bounds__(256) void dp_render_naive(
    const float* __restrict__ rays_o,
    const float* __restrict__ rays_d,
    const float* __restrict__ grid,
    const float* __restrict__ atoms,
    float* __restrict__ out)
{
    __shared__ float AL[8][NATOMS];
    const int tid = threadIdx.x;
    for (int i = tid; i < 8 * NATOMS; i += 256) {
        AL[i >> 5][i & 31] = atoms[i * DDIM + (DDIM - 1)];
    }
    __syncthreads();

    const int r = blockIdx.x;
    const int s = tid;
    if (s >= NSAMP) return;

    const float ox = rays_o[r * 3 + 0], oy = rays_o[r * 3 + 1], oz = rays_o[r * 3 + 2];
    const float dx = rays_d[r * 3 + 0], dy = rays_d[r * 3 + 1], dz = rays_d[r * 3 + 2];

    const float tx = fminf((1.0f - ox) / dx, (-1.0f - ox) / dx);
    const float ty = fminf((1.0f - oy) / dy, (-1.0f - oy) / dy);
    const float tz = fminf((1.0f - oz) / dz, (-1.0f - oz) / dz);
    const float start = fmaxf(fmaxf(tx, ty), tz);

    const float t  = start + (float)s * STEP_F;
    const float px = ox + t * dx;
    const float py = oy + t * dy;
    const float pz = oz + t * dz;

    const bool mask = (px > -1.0f) && (px < 1.0f) &&
                      (py > -1.0f) && (py < 1.0f) &&
                      (pz > -1.0f) && (pz < 1.0f);

    const float gx = (px + 1.0f) * (1.0f / VOXEL_F);
    const float gy = (py + 1.0f) * (1.0f / VOXEL_F);
    const float gz = (pz + 1.0f) * (1.0f / VOXEL_F);

    float acc = 0.0f;
    for (int n = 0; n < 8; ++n) {
        const float cx = (n & 4) ? 0.5f : -0.5f;
        const float cy = (n & 2) ? 0.5f : -0.5f;
        const float cz = (n & 1) ? 0.5f : -0.5f;

        float fx = fminf(fmaxf(floorf(gx + cx), 0.0f), (float)(GRES - 1));
        float fy = fminf(fmaxf(floorf(gy + cy), 0.0f), (float)(GRES - 1));
        float fz = fminf(fmaxf(floorf(gz + cz), 0.0f), (float)(GRES - 1));

        const float wx = 1.0f - fabsf(gx - (fx + 0.5f));
        const float wy = 1.0f - fabsf(gy - (fy + 0.5f));
        const float wz = 1.0f - fabsf(gz - (fz + 0.5f));
        const float w  = wx * wy * wz;

        const int ix = (int)fx, iy = (int)fy, iz = (int)fz;
        const int cell = ((ix >> 1) * 32 + (iy >> 1)) * 32 + (iz >> 1);
        const int f    = ((ix & 1) << 2) | ((iy & 1) << 1) | (iz & 1);

        const float* __restrict__ grow = grid + (size_t)cell * NATOMS;
        float dp = 0.0f;
#pragma unroll
        for (int a = 0; a < NATOMS; ++a) dp = fmaf(grow[a], AL[f][a], dp);
        acc = fmaf(w, dp, acc);
    }

    out[r * NSAMP + s] = mask ? fmaxf(acc, 0.0f) : 0.0f;
}

// ---------------------------------------------------------------------------
extern "C" void kernel_launch(void* const* d_in, const int* in_sizes, int n_in,
                              void* d_out, int out_size, void* d_ws, size_t ws_size,
                              hipStream_t stream)
{
    const float* rays_o = (const float*)d_in[0];
    const float* rays_d = (const float*)d_in[1];
    const float* grid   = (const float*)d_in[2];
    const float* atoms  = (const float*)d_in[3];
    // d_in[4] = grid_id (unused by the computation)
    float* out = (float*)d_out;

    const size_t S_bytes = (size_t)NCELL * 8 * sizeof(float);  // 1 MB

    if (ws_size >= S_bytes) {
        float* S = (float*)d_ws;
        dp_precompute_S<<<256, 256, 0, stream>>>(grid, atoms, S);
        dp_render<<<NRAYS, 256, 0, stream>>>(rays_o, rays_d, S, out);
    } else {
        dp_render_naive<<<NRAYS, 256, 0, stream>>>(rays_o, rays_d, grid, atoms, out);
    }
}